// Encoder_65395172049150
// MI455X (gfx1250) — compile-verified
//
#include <hip/hip_runtime.h>

// ---------------------------------------------------------------------------
// Transformer encoder layer (fp32) for gfx1250 using V_WMMA_F32_16X16X4_F32.
// Outputs: x2 [B,S,512] followed by heatmap (raw pre-mask scores) [H*B,S,S].
// ---------------------------------------------------------------------------

#define S_LEN  2048
#define BATCH  2
#define DMODEL 512
#define DHID   2048
#define HEADS  8
#define HDIM   64
#define NROWS  (BATCH * S_LEN)        // 4096

typedef float v2f __attribute__((ext_vector_type(2)));
typedef float v8f __attribute__((ext_vector_type(8)));

__device__ __forceinline__ v8f wmma4(v2f a, v2f b, v8f c) {
  // D = A(16x4,f32) x B(4x16,f32) + C(16x16,f32)
  return __builtin_amdgcn_wmma_f32_16x16x4_f32(false, a, false, b, (short)0, c,
                                               false, false);
}

// ---------------------------------------------------------------------------
// Block reductions (LDS)
// ---------------------------------------------------------------------------
__device__ __forceinline__ float block_reduce_sum(float v) {
  __shared__ float red[256];
  const int t = threadIdx.x;
  red[t] = v;
  __syncthreads();
  for (int s = 128; s > 0; s >>= 1) {
    if (t < s) red[t] += red[t + s];
    __syncthreads();
  }
  float r = red[0];
  __syncthreads();
  return r;
}

__device__ __forceinline__ float block_reduce_max(float v) {
  __shared__ float redm[256];
  const int t = threadIdx.x;
  redm[t] = v;
  __syncthreads();
  for (int s = 128; s > 0; s >>= 1) {
    if (t < s) redm[t] = fmaxf(redm[t], redm[t + s]);
    __syncthreads();
  }
  float r = redm[0];
  __syncthreads();
  return r;
}

// ---------------------------------------------------------------------------
// Generic WMMA GEMM: C = A[MxK] @ B[KxN] + bias, 256 threads = 8 waves,
// block tile 128x64, wave tile 32x32 (2x2 WMMA 16x16 accumulators).
// MODE 1: head-major scatter store (QKV projections)
// MODE 2: + residual add, row-major store
// MODE 3: + ReLU, row-major store
// ---------------------------------------------------------------------------
template <int MODE>
__global__ __launch_bounds__(256) void gemm_kernel(
    const float* __restrict__ A, const float* __restrict__ Bm,
    const float* __restrict__ bias, const float* __restrict__ res,
    float* __restrict__ C, int M, int N, int K) {
  const int wave = threadIdx.x >> 5;
  const int lane = threadIdx.x & 31;
  const int half = lane >> 4;
  const int lm   = lane & 15;
  const int mbase = blockIdx.x * 128 + (wave & 3) * 32;
  const int nbase = blockIdx.y * 64 + (wave >> 2) * 32;

  v8f acc[2][2] = {};
  for (int kk = 0; kk < K; kk += 4) {
    const int ka = kk + 2 * half;
    v2f a[2], b[2];
#pragma unroll
    for (int mi = 0; mi < 2; ++mi)
      a[mi] = *(const v2f*)(A + (size_t)(mbase + mi * 16 + lm) * K + ka);
#pragma unroll
    for (int ni = 0; ni < 2; ++ni) {
      const float* bp = Bm + (size_t)ka * N + (nbase + ni * 16 + lm);
      b[ni].x = bp[0];
      b[ni].y = bp[N];
    }
#pragma unroll
    for (int mi = 0; mi < 2; ++mi)
#pragma unroll
      for (int ni = 0; ni < 2; ++ni)
        acc[mi][ni] = wmma4(a[mi], b[ni], acc[mi][ni]);
  }

#pragma unroll
  for (int mi = 0; mi < 2; ++mi) {
#pragma unroll
    for (int ni = 0; ni < 2; ++ni) {
      const int col = nbase + ni * 16 + lm;
      const float bv = bias[col];
#pragma unroll
      for (int r = 0; r < 8; ++r) {
        const int row = mbase + mi * 16 + r + 8 * half;
        float v = acc[mi][ni][r] + bv;
        if (MODE == 2) v += res[(size_t)row * N + col];
        if (MODE == 3) v = fmaxf(v, 0.0f);
        if (MODE == 1) {
          const int h = col >> 6, d = col & 63;       // head, dim
          const int bb = row >> 11, ss = row & (S_LEN - 1);
          C[(((size_t)(h * BATCH + bb)) * S_LEN + ss) * HDIM + d] = v;
        } else {
          C[(size_t)row * N + col] = v;
        }
      }
    }
  }
}

// ---------------------------------------------------------------------------
// Scores: P[s][t] = sum_d Q[s][d] * K[t][d] per (h,b); raw scores -> heatmap.
// ---------------------------------------------------------------------------
__global__ __launch_bounds__(256) void scores_kernel(
    const float* __restrict__ Q, const float* __restrict__ Kh,
    float* __restrict__ heat) {
  const int hb = blockIdx.z;
  const float* q = Q + (size_t)hb * S_LEN * HDIM;
  const float* k = Kh + (size_t)hb * S_LEN * HDIM;
  float* out = heat + (size_t)hb * S_LEN * S_LEN;

  const int wave = threadIdx.x >> 5;
  const int lane = threadIdx.x & 31;
  const int half = lane >> 4;
  const int lm   = lane & 15;
  const int mbase = blockIdx.x * 128 + (wave & 3) * 32;
  const int nbase = blockIdx.y * 64 + (wave >> 2) * 32;

  v8f acc[2][2] = {};
#pragma unroll
  for (int kk = 0; kk < HDIM; kk += 4) {
    const int ka = kk + 2 * half;
    v2f a[2], b[2];
#pragma unroll
    for (int mi = 0; mi < 2; ++mi)
      a[mi] = *(const v2f*)(q + (size_t)(mbase + mi * 16 + lm) * HDIM + ka);
#pragma unroll
    for (int ni = 0; ni < 2; ++ni)   // B = K^T: B[k][t] = K[t][k]
      b[ni] = *(const v2f*)(k + (size_t)(nbase + ni * 16 + lm) * HDIM + ka);
#pragma unroll
    for (int mi = 0; mi < 2; ++mi)
#pragma unroll
      for (int ni = 0; ni < 2; ++ni)
        acc[mi][ni] = wmma4(a[mi], b[ni], acc[mi][ni]);
  }
#pragma unroll
  for (int mi = 0; mi < 2; ++mi)
#pragma unroll
    for (int ni = 0; ni < 2; ++ni) {
      const int col = nbase + ni * 16 + lm;
#pragma unroll
      for (int r = 0; r < 8; ++r) {
        const int row = mbase + mi * 16 + r + 8 * half;
        out[(size_t)row * S_LEN + col] = acc[mi][ni][r];
      }
    }
}

// ---------------------------------------------------------------------------
// Per-row causal softmax stats: max over t<=s and 1/sum(exp(p-max)).
// One 256-thread block per row (HB*S rows).
// ---------------------------------------------------------------------------
__global__ __launch_bounds__(256) void softmax_stats_kernel(
    const float* __restrict__ heat, float* __restrict__ stats) {
  const int row = blockIdx.x;                 // hb*S + s
  const int hb = row >> 11;
  const int s  = row & (S_LEN - 1);
  const float* p = heat + (size_t)hb * S_LEN * S_LEN + (size_t)s * S_LEN;

  float mx = -3.0e38f;
  for (int t = threadIdx.x; t <= s; t += 256) mx = fmaxf(mx, p[t]);
  mx = block_reduce_max(mx);

  float sum = 0.0f;
  for (int t = threadIdx.x; t <= s; t += 256) sum += __expf(p[t] - mx);
  sum = block_reduce_sum(sum);

  if (threadIdx.x == 0) {
    stats[(size_t)row * 2 + 0] = mx;
    stats[(size_t)row * 2 + 1] = 1.0f / sum;
  }
}

// ---------------------------------------------------------------------------
// PV: wv[s][:] = softmax_row(s) @ V, softmax rebuilt on the fly from heatmap.
// Causal early-exit: k-blocks beyond the wave's last row are all zero.
// Output scattered to [B,S,H*64] concatenated-head layout.
// ---------------------------------------------------------------------------
__global__ __launch_bounds__(256) void pv_kernel(
    const float* __restrict__ heat, const float* __restrict__ stats,
    const float* __restrict__ V, float* __restrict__ wv) {
  const int hb = blockIdx.z;
  const int h  = hb / BATCH;
  const int bb = hb % BATCH;
  const float* sc = heat + (size_t)hb * S_LEN * S_LEN;
  const float* vv = V + (size_t)hb * S_LEN * HDIM;

  const int wave = threadIdx.x >> 5;
  const int lane = threadIdx.x & 31;
  const int half = lane >> 4;
  const int lm   = lane & 15;
  const int mbase = blockIdx.x * 128 + (wave & 3) * 32;
  const int nbase = (wave >> 2) * 32;          // N = 64 total

  int   rows[2];
  float mx[2], inv[2];
#pragma unroll
  for (int mi = 0; mi < 2; ++mi) {
    rows[mi] = mbase + mi * 16 + lm;
    mx[mi]  = stats[((size_t)hb * S_LEN + rows[mi]) * 2 + 0];
    inv[mi] = stats[((size_t)hb * S_LEN + rows[mi]) * 2 + 1];
  }

  v8f acc[2][2] = {};
  const int kend = mbase + 32;                 // causal: k > last row is masked
  for (int kk = 0; kk < kend; kk += 4) {
    const int ka = kk + 2 * half;
    v2f a[2], b[2];
#pragma unroll
    for (int mi = 0; mi < 2; ++mi) {
      v2f s2 = *(const v2f*)(sc + (size_t)rows[mi] * S_LEN + ka);
      a[mi].x = (ka     <= rows[mi]) ? __expf(s2.x - mx[mi]) * inv[mi] : 0.0f;
      a[mi].y = (ka + 1 <= rows[mi]) ? __expf(s2.y - mx[mi]) * inv[mi] : 0.0f;
    }
#pragma unroll
    for (int ni = 0; ni < 2; ++ni) {
      const float* bp = vv + (size_t)ka * HDIM + (nbase + ni * 16 + lm);
      b[ni].x = bp[0];
      b[ni].y = bp[HDIM];
    }
#pragma unroll
    for (int mi = 0; mi < 2; ++mi)
#pragma unroll
      for (int ni = 0; ni < 2; ++ni)
        acc[mi][ni] = wmma4(a[mi], b[ni], acc[mi][ni]);
  }

#pragma unroll
  for (int mi = 0; mi < 2; ++mi)
#pragma unroll
    for (int ni = 0; ni < 2; ++ni) {
      const int col = nbase + ni * 16 + lm;
#pragma unroll
      for (int r = 0; r < 8; ++r) {
        const int row = mbase + mi * 16 + r + 8 * half;
        wv[((size_t)bb * S_LEN + row) * DMODEL + h * HDIM + col] = acc[mi][ni][r];
      }
    }
}

// ---------------------------------------------------------------------------
// LayerNorm over D=512; one 256-thread block per row, 2 elements/thread.
// ---------------------------------------------------------------------------
__global__ __launch_bounds__(256) void layernorm_kernel(
    const float* __restrict__ in, const float* __restrict__ g,
    const float* __restrict__ bta, float* __restrict__ out) {
  const int row = blockIdx.x;
  const int t = threadIdx.x;
  const float* p = in + (size_t)row * DMODEL;
  const float x0 = p[t];
  const float x1 = p[t + 256];
  const float mu = block_reduce_sum(x0 + x1) * (1.0f / DMODEL);
  const float d0 = x0 - mu, d1 = x1 - mu;
  const float var = block_reduce_sum(d0 * d0 + d1 * d1) * (1.0f / DMODEL);
  const float rs = 1.0f / sqrtf(var + 1e-5f);
  out[(size_t)row * DMODEL + t]       = d0 * rs * g[t] + bta[t];
  out[(size_t)row * DMODEL + t + 256] = d1 * rs * g[t + 256] + bta[t + 256];
}

// ---------------------------------------------------------------------------
// Launch
// ---------------------------------------------------------------------------
extern "C" void kernel_launch(void* const* d_in, const int* in_sizes, int n_in,
                              void* d_out, int out_size, void* d_ws,
                              size_t ws_size, hipStream_t stream) {
  const float* x   = (const float*)d_in[0];
  const float* Wq  = (const float*)d_in[1];
  const float* bq  = (const float*)d_in[2];
  const float* Wk  = (const float*)d_in[3];
  const float* bk  = (const float*)d_in[4];
  const float* Wv  = (const float*)d_in[5];
  const float* bv  = (const float*)d_in[6];
  const float* Wo  = (const float*)d_in[7];
  const float* bo  = (const float*)d_in[8];
  const float* W1  = (const float*)d_in[9];
  const float* b1  = (const float*)d_in[10];
  const float* W2  = (const float*)d_in[11];
  const float* b2  = (const float*)d_in[12];
  const float* g1  = (const float*)d_in[13];
  const float* be1 = (const float*)d_in[14];
  const float* g2  = (const float*)d_in[15];
  const float* be2 = (const float*)d_in[16];

  float* out_x2 = (float*)d_out;
  float* heat   = (float*)d_out + (size_t)NROWS * DMODEL;   // [H*B,S,S]

  // Workspace layout (floats)
  float* ws    = (float*)d_ws;
  float* Q     = ws;                               // 4096*512
  float* Kw    = Q  + (size_t)NROWS * DMODEL;
  float* Vw    = Kw + (size_t)NROWS * DMODEL;
  float* stats = Vw + (size_t)NROWS * DMODEL;      // 16*2048*2 = 65536
  float* wv    = stats + (size_t)HEADS * BATCH * S_LEN * 2;
  float* tmp   = wv  + (size_t)NROWS * DMODEL;
  float* x1    = tmp + (size_t)NROWS * DMODEL;
  float* h1    = x1  + (size_t)NROWS * DMODEL;     // 4096*2048
  float* fftmp = h1  + (size_t)NROWS * DHID;

  const dim3 blk(256);

  // QKV projections (head-major outputs)
  gemm_kernel<1><<<dim3(32, 8), blk, 0, stream>>>(x, Wq, bq, nullptr, Q,
                                                  NROWS, DMODEL, DMODEL);
  gemm_kernel<1><<<dim3(32, 8), blk, 0, stream>>>(x, Wk, bk, nullptr, Kw,
                                                  NROWS, DMODEL, DMODEL);
  gemm_kernel<1><<<dim3(32, 8), blk, 0, stream>>>(x, Wv, bv, nullptr, Vw,
                                                  NROWS, DMODEL, DMODEL);
  // Raw scores -> heatmap output region
  scores_kernel<<<dim3(16, 32, 16), blk, 0, stream>>>(Q, Kw, heat);
  // Causal softmax row statistics
  softmax_stats_kernel<<<dim3(16 * S_LEN), blk, 0, stream>>>(heat, stats);
  // attn @ V with on-the-fly softmax, scatter to [B,S,H*64]
  pv_kernel<<<dim3(16, 1, 16), blk, 0, stream>>>(heat, stats, Vw, wv);
  // Output projection + residual
  gemm_kernel<2><<<dim3(32, 8), blk, 0, stream>>>(wv, Wo, bo, x, tmp,
                                                  NROWS, DMODEL, DMODEL);
  layernorm_kernel<<<dim3(NROWS), blk, 0, stream>>>(tmp, g1, be1, x1);
  // FFN
  gemm_kernel<3><<<dim3(32, 32), blk, 0, stream>>>(x1, W1, b1, nullptr, h1,
                                                   NROWS, DHID, DMODEL);
  gemm_kernel<2><<<dim3(32, 8), blk, 0, stream>>>(h1, W2, b2, x1, fftmp,
                                                  NROWS, DMODEL, DHID);
  layernorm_kernel<<<dim3(NROWS), blk, 0, stream>>>(fftmp, g2, be2, out_x2);
}